// MultiHeadAttention_8684423872640
// MI455X (gfx1250) — compile-verified
//
#include <hip/hip_runtime.h>

#define BB 8
#define LL 4096
#define NCOL 1024      // N_IN == N_LATENT
#define NHEADS 16
#define DK 64
#define CHUNKS 32
#define ROWS_PER_CHUNK (LL / CHUNKS)   // 128
#define KWAVES 8
#define KSLICE (NCOL / KWAVES)         // 128 k per wave

typedef __attribute__((ext_vector_type(2))) float v2f;
typedef __attribute__((ext_vector_type(8))) float v8f;

// ---------------- Stage 1a: partial column sums of x per (batch, chunk) ----------------
__global__ void col_partial_sum(const float* __restrict__ x, float* __restrict__ partial) {
    int chunk = blockIdx.x;            // 0..31
    int b     = blockIdx.y;            // 0..7
    int n4    = threadIdx.x;           // 0..255 -> columns n4*4..n4*4+3
    const float4* xr = (const float4*)(x + (size_t)b * LL * NCOL);
    float4 acc = make_float4(0.f, 0.f, 0.f, 0.f);
    int l0 = chunk * ROWS_PER_CHUNK;
    for (int l = l0; l < l0 + ROWS_PER_CHUNK; ++l) {
        const float4* p = &xr[(size_t)l * (NCOL / 4) + n4];
        __builtin_prefetch((const void*)(p + 4 * (NCOL / 4)), 0, 0); // global_prefetch_b8
        float4 vv = *p;
        acc.x += vv.x; acc.y += vv.y; acc.z += vv.z; acc.w += vv.w;
    }
    ((float4*)partial)[((size_t)(b * CHUNKS + chunk)) * (NCOL / 4) + n4] = acc;
}

// ---------------- Stage 1b: reduce chunks -> xbarP (padded to 16 rows, rows 8..15 = 0) ----
__global__ void col_mean_finalize(const float* __restrict__ partial, float* __restrict__ xbarP) {
    int b  = blockIdx.x;               // 0..15 (padded)
    int n4 = threadIdx.x;
    float4 r = make_float4(0.f, 0.f, 0.f, 0.f);
    if (b < BB) {
        float4 acc = make_float4(0.f, 0.f, 0.f, 0.f);
        for (int c = 0; c < CHUNKS; ++c) {
            float4 vv = ((const float4*)partial)[((size_t)(b * CHUNKS + c)) * (NCOL / 4) + n4];
            acc.x += vv.x; acc.y += vv.y; acc.z += vv.z; acc.w += vv.w;
        }
        const float s = 1.0f / (float)LL;
        r = make_float4(acc.x * s, acc.y * s, acc.z * s, acc.w * s);
    }
    ((float4*)xbarP)[(size_t)b * (NCOL / 4) + n4] = r;
}

// ---------------- Stage 2: vP = (Wq @ xbarP^T + bq) * Wfc[h]/8  via f32 WMMA ----------------
// GEMM: A = Wq [1024x1024], B = xbarP^T [1024x16], D [1024x16].
// 64 blocks (one per 16-row M-tile) x 8 waves splitting K; LDS cross-wave reduce.
__global__ void qbar_v_wmma(const float* __restrict__ Wq, const float* __restrict__ bq,
                            const float* __restrict__ Wfc, const float* __restrict__ xbarP,
                            float* __restrict__ vP) {
    __shared__ float red[KWAVES][32][9];   // padded stride 9 -> no bank conflicts
    int tid   = threadIdx.x;
    int wave  = tid >> 5;
    int lane  = tid & 31;
    int m0    = blockIdx.x * 16;
    int row   = m0 + (lane & 15);        // A row for this lane
    int khalf = (lane >> 4) * 2;         // lanes 0-15: K=0,1 ; lanes 16-31: K=2,3
    int col   = lane & 15;               // B/D column (batch, padded)
    int k0    = wave * KSLICE;
    v8f acc = {};
    for (int k = k0; k < k0 + KSLICE; k += 4) {
        v2f a, bm;
        a.x  = Wq[(size_t)row * NCOL + k + khalf];
        a.y  = Wq[(size_t)row * NCOL + k + khalf + 1];
        bm.x = xbarP[(size_t)col * NCOL + k + khalf];
        bm.y = xbarP[(size_t)col * NCOL + k + khalf + 1];
        acc = __builtin_amdgcn_wmma_f32_16x16x4_f32(false, a, false, bm,
                                                    (short)0, acc, false, false);
    }
    #pragma unroll
    for (int j = 0; j < 8; ++j) red[wave][lane][j] = acc[j];
    __syncthreads();
    if (wave == 0) {
        v8f tot = {};
        #pragma unroll
        for (int w = 0; w < KWAVES; ++w)
            #pragma unroll
            for (int j = 0; j < 8; ++j) tot[j] += red[w][lane][j];
        // D layout: VGPR j -> (lanes 0-15: M=m0+j) (lanes 16-31: M=m0+8+j), N=col
        int mbase = m0 + ((lane >> 4) ? 8 : 0);
        #pragma unroll
        for (int j = 0; j < 8; ++j) {
            int d = mbase + j;
            float q = tot[j] + bq[d];
            float val = (col < BB) ? q * Wfc[d / DK] * 0.125f : 0.0f; // pad rows -> 0
            vP[(size_t)col * NCOL + d] = val;
        }
    }
}

// ---------------- Stage 3: w_eff = vP @ Wk  via f32 WMMA ----------------
// GEMM: A = vP [16x1024] (rows 8..15 zero), B = Wk [1024 x 16-tile], D [16x16] per N-tile.
__global__ void weff_wmma(const float* __restrict__ Wk, const float* __restrict__ vP,
                          float* __restrict__ weff) {
    __shared__ float red[KWAVES][32][9];
    int tid   = threadIdx.x;
    int wave  = tid >> 5;
    int lane  = tid & 31;
    int n0    = blockIdx.x * 16;
    int m     = lane & 15;               // batch row of A (zero-padded)
    int col   = lane & 15;               // B/D column
    int khalf = (lane >> 4) * 2;
    int k0    = wave * KSLICE;
    v8f acc = {};
    for (int k = k0; k < k0 + KSLICE; k += 4) {
        v2f a, bm;
        a.x  = vP[(size_t)m * NCOL + k + khalf];
        a.y  = vP[(size_t)m * NCOL + k + khalf + 1];
        bm.x = Wk[(size_t)(k + khalf)     * NCOL + n0 + col];
        bm.y = Wk[(size_t)(k + khalf + 1) * NCOL + n0 + col];
        acc = __builtin_amdgcn_wmma_f32_16x16x4_f32(false, a, false, bm,
                                                    (short)0, acc, false, false);
    }
    #pragma unroll
    for (int j = 0; j < 8; ++j) red[wave][lane][j] = acc[j];
    __syncthreads();
    if (wave == 0 && lane < 16) {
        // D: lanes 0-15 hold M=0..7 (real batches) across acc VGPRs 0..7
        #pragma unroll
        for (int j = 0; j < 8; ++j) {
            float tot = 0.f;
            #pragma unroll
            for (int w = 0; w < KWAVES; ++w) tot += red[w][lane][j];
            weff[(size_t)j * NCOL + n0 + col] = tot;
        }
    }
}

// ---------------- Stage 4: c[b] = dot(vP[b], bk) + bfc ----------------
__global__ void c_reduce(const float* __restrict__ vP, const float* __restrict__ bk,
                         const float* __restrict__ bfc, float* __restrict__ cvec) {
    int b    = threadIdx.x >> 5;   // 8 waves, one per batch
    int lane = threadIdx.x & 31;
    float acc = 0.f;
    for (int d = lane; d < NCOL; d += 32)
        acc += vP[(size_t)b * NCOL + d] * bk[d];
    for (int off = 16; off > 0; off >>= 1)
        acc += __shfl_xor(acc, off, 32);
    if (lane == 0) cvec[b] = acc + bfc[0];
}

// ---------------- Stage 5: out[b,l] = x[b,l,:].w_eff[b] + c[b] (streaming pass) ----------------
__global__ void out_dot(const float* __restrict__ x, const float* __restrict__ weff,
                        const float* __restrict__ cvec, float* __restrict__ out) {
    int wave = threadIdx.x >> 5;                       // 8 waves/block
    int lane = threadIdx.x & 31;
    size_t row = (size_t)blockIdx.x * 8 + wave;        // 0..32767
    int b = (int)(row >> 12);                          // row / 4096
    const float4* xr = (const float4*)(x + row * NCOL);
    const float4* wr = (const float4*)(weff + (size_t)b * NCOL);
    float acc = 0.f;
    #pragma unroll
    for (int i = 0; i < 8; ++i) {
        float4 xv = xr[i * 32 + lane];
        float4 wv = wr[i * 32 + lane];
        acc += xv.x * wv.x + xv.y * wv.y + xv.z * wv.z + xv.w * wv.w;
    }
    for (int off = 16; off > 0; off >>= 1)
        acc += __shfl_xor(acc, off, 32);
    if (lane == 0) out[row] = acc + cvec[b];
}

extern "C" void kernel_launch(void* const* d_in, const int* in_sizes, int n_in,
                              void* d_out, int out_size, void* d_ws, size_t ws_size,
                              hipStream_t stream) {
    const float* x   = (const float*)d_in[0];
    const float* Wq  = (const float*)d_in[1];
    const float* bq  = (const float*)d_in[2];
    const float* Wk  = (const float*)d_in[3];
    const float* bk  = (const float*)d_in[4];
    const float* Wfc = (const float*)d_in[5];
    const float* bfc = (const float*)d_in[6];
    float* out = (float*)d_out;

    float* ws      = (float*)d_ws;
    float* partial = ws;                                   // 8*32*1024
    float* xbarP   = partial + (size_t)BB * CHUNKS * NCOL; // 16*1024 (padded)
    float* vP      = xbarP + (size_t)16 * NCOL;            // 16*1024 (padded)
    float* weff    = vP + (size_t)16 * NCOL;               // 8*1024
    float* cvec    = weff + (size_t)BB * NCOL;             // 8

    col_partial_sum  <<<dim3(CHUNKS, BB), 256, 0, stream>>>(x, partial);
    col_mean_finalize<<<16, 256, 0, stream>>>(partial, xbarP);
    qbar_v_wmma      <<<NCOL / 16, 256, 0, stream>>>(Wq, bq, Wfc, xbarP, vP);
    weff_wmma        <<<NCOL / 16, 256, 0, stream>>>(Wk, vP, weff);
    c_reduce         <<<1, 256, 0, stream>>>(vP, bk, bfc, cvec);
    out_dot          <<<(BB * LL) / 8, 256, 0, stream>>>(x, weff, cvec, out);
}